// GRAMModel_27805618275293
// MI455X (gfx1250) — compile-verified
//
#include <hip/hip_runtime.h>
#include <hip/hip_bf16.h>

typedef __attribute__((ext_vector_type(2))) float v2f;
typedef __attribute__((ext_vector_type(8))) float v8f;

#define H 128
#define P_DIM 16
#define B_DIM 4096
#define T_DIM 262144

// ---- workspace layout (float offsets) ----
// [all_emb][le][re][wsum][aw]   <- le..aw zeroed contiguously each launch
// [Wt packed][bilin][Wl packed]
#define OFF_ALLEMB 0u
#define N_ALLEMB   (18000u * 128u)                 // 2,304,000
#define OFF_LE     (OFF_ALLEMB + N_ALLEMB)
#define OFF_RE     (OFF_LE + 4096u * 128u)
#define OFF_W      (OFF_RE + 4096u * 128u)         // 3 groups x 8
#define OFF_AW     (OFF_W + 32u)                   // 40000+16000+20000 = 76000
#define OFF_WT     (OFF_AW + 76000u)               // W_ntn K-pair packed: 16*64*128*2
#define OFF_BILIN  (OFF_WT + 262144u)              // 4096*16
#define OFF_WLPK   (OFF_BILIN + 65536u)            // 3 x 128*128*2
#define NZERO      (4096u * 128u * 2u + 32u + 76000u)

// ------------------------------------------------------------------
__global__ void zero_f32(float* __restrict__ p, unsigned n) {
  unsigned i = blockIdx.x * blockDim.x + threadIdx.x;
  if (i < n) p[i] = 0.f;
}

// Wl (k=128 x c=256) -> K-pair pack: out[cp*256 + k*2 + s] = Wl[k*256 + 2*cp + s]
// Lane fragment {B[c][k], B[c+1][k]} becomes one aligned float2.
__global__ void pack_wl(const float* __restrict__ Wl, float* __restrict__ out) {
  unsigned i = blockIdx.x * blockDim.x + threadIdx.x;
  if (i >= 128u * 256u) return;
  unsigned cp = i >> 8, k = (i & 255u) >> 1, s = i & 1u;
  out[i] = Wl[k * 256u + 2u * cp + s];
}

// W_ntn (i,j,p) -> out[p*16384 + kp*256 + j*2 + s] = W[(2kp+s)*2048 + j*16 + p]
__global__ void pack_wntn(const float* __restrict__ W, float* __restrict__ out) {
  unsigned i = blockIdx.x * blockDim.x + threadIdx.x;
  if (i >= 128u * 128u * 16u) return;
  unsigned p = i >> 14, r = i & 16383u;
  unsigned kp = r >> 8, j = (r & 255u) >> 1, s = i & 1u;
  out[i] = W[(2u * kp + s) * 2048u + j * 16u + p];
}

// DAG attention logits on the fp32 matrix path.
// Grid (row_tiles, 8 k-tiles), 1 wave. D(16row x 16k) = x(16x256) @ Wl^T tile,
// epilogue tanh(D + bl[k]) * ap[k], reduce over k cols, atomicAdd into aw[row].
__global__ void attn_wmma(const int* __restrict__ anc, const int* __restrict__ leaf,
                          const float* __restrict__ emb, const float* __restrict__ Wlpk,
                          const float* __restrict__ bl, const float* __restrict__ ap,
                          float* __restrict__ aw) {
  int tile = blockIdx.x;
  int kt   = blockIdx.y;               // 0..7
  int lane = threadIdx.x;              // full wave, EXEC all-1s
  int half = lane >> 4, lr = lane & 15;
  int row  = tile * 16 + lr;
  const float* pA = emb + anc[row] * H;   // per-lane gathered row base
  const float* pL = emb + leaf[row] * H;
  int kcol = kt * 16 + lr;
  float blk = bl[kcol], apk = ap[kcol];
  const float* Bb = Wlpk + (unsigned)kcol * 2u;

  v8f C = {};
  #pragma unroll 4
  for (int kc = 0; kc < 64; ++kc) {     // K = 256 over c
    int c0 = kc * 4;
    const float* xp = (c0 < 128) ? (pA + c0) : (pL + (c0 - 128));
    v2f A  = *(const v2f*)(xp + 2 * half);                    // b64, 8B aligned
    v2f Bv = *(const v2f*)(Bb + (unsigned)(kc * 2 + half) * 256u);
    C = __builtin_amdgcn_wmma_f32_16x16x4_f32(false, A, false, Bv,
                                              (short)0, C, false, false);
  }
  #pragma unroll
  for (int r = 0; r < 8; ++r) {
    float v = tanhf(C[r] + blk) * apk;  // lane owns column kcol, row m
    for (int off = 1; off < 16; off <<= 1) v += __shfl_xor(v, off, 32);
    if (lr == 0) {
      int m = half ? (r + 8) : r;
      atomicAdd(&aw[tile * 16 + m], v); // partial over this 16-k slice
    }
  }
}

// Per-v softmax over L logits, accumulate into the L global weights
__global__ void attn_softmax(const float* __restrict__ aw, float* __restrict__ wsum,
                             int V, int L) {
  int v = blockIdx.x * blockDim.x + threadIdx.x;
  if (v >= V) return;
  float a[8];
  float m = -1e30f;
  for (int l = 0; l < L; ++l) { a[l] = aw[v * L + l]; m = fmaxf(m, a[l]); }
  float s = 0.f;
  for (int l = 0; l < L; ++l) { a[l] = expf(a[l] - m); s += a[l]; }
  float inv = 1.f / s;
  for (int l = 0; l < L; ++l) atomicAdd(&wsum[l], a[l] * inv);
}

// all_emb[v,c] = sum_l wsum[l] * emb[anc[v,l], c]
__global__ void emb_combine(const int* __restrict__ anc, const float* __restrict__ emb,
                            const float* __restrict__ wsum, float* __restrict__ out, int L) {
  int v = blockIdx.x, c = threadIdx.x;
  float acc = 0.f;
  for (int l = 0; l < L; ++l) acc += wsum[l] * emb[anc[v * L + l] * H + c];
  out[v * H + c] = acc;
}

// Sorted segment-sum: accumulate runs in registers, flush on segment change.
__global__ void seg_sum(const int* __restrict__ xidx, const int* __restrict__ xbatch,
                        const float* __restrict__ all_emb, float* __restrict__ dst) {
  const int CHUNK = 64;
  int t0 = blockIdx.x * CHUNK;
  int c = threadIdx.x;                 // 0..127
  int cur = xbatch[t0];
  float acc = 0.f;
  for (int t = t0; t < t0 + CHUNK; ++t) {
    int b = xbatch[t];
    if (b != cur) {
      atomicAdd(&dst[cur * H + c], acc);
      acc = 0.f; cur = b;
    }
    acc += all_emb[xidx[t] * H + c];   // coalesced 512B row gather (L2-resident)
  }
  atomicAdd(&dst[cur * H + c], acc);
}

// Fused bilinear: one wave per (16-row b-tile, p). 8 j-tile accumulators live
// in registers so each A load feeds 8 WMMAs; B loads are packed b64.
__global__ void bilinear_wmma(const float* __restrict__ leS, const float* __restrict__ reS,
                              const float* __restrict__ Wpk, float* __restrict__ bilin) {
  int bt = blockIdx.x;                 // 0..255
  int p  = blockIdx.y;                 // 0..15
  int lane = threadIdx.x;
  int half = lane >> 4, lr = lane & 15;
  int b0 = bt * 16;
  const float* leRow = leS + (b0 + lr) * H + 2 * half;
  const float* Bb = Wpk + (unsigned)p * 16384u + (unsigned)lr * 2u;

  v8f C[8];
  #pragma unroll
  for (int jt = 0; jt < 8; ++jt) C[jt] = (v8f){};

  #pragma unroll 2
  for (int kc = 0; kc < 32; ++kc) {    // K = i (128)
    v2f A = *(const v2f*)(leRow + kc * 4);                    // rows of le tile
    const float* bp = Bb + (unsigned)(kc * 2 + half) * 256u;
    #pragma unroll
    for (int jt = 0; jt < 8; ++jt) {
      v2f Bv = *(const v2f*)(bp + jt * 32);                   // packed b64
      C[jt] = __builtin_amdgcn_wmma_f32_16x16x4_f32(false, A, false, Bv,
                                                    (short)0, C[jt], false, false);
    }
  }
  // epilogue: (le @ W_p) elementwise * re, reduce over j
  float dotacc[8];
  #pragma unroll
  for (int r = 0; r < 8; ++r) dotacc[r] = 0.f;
  #pragma unroll
  for (int jt = 0; jt < 8; ++jt) {
    #pragma unroll
    for (int r = 0; r < 8; ++r) {
      int m = half ? (r + 8) : r;
      dotacc[r] += C[jt][r] * reS[(b0 + m) * H + jt * 16 + lr];
    }
  }
  #pragma unroll
  for (int r = 0; r < 8; ++r) {
    float v = dotacc[r];
    for (int off = 1; off < 16; off <<= 1) v += __shfl_xor(v, off, 32);
    if (lr == 0) {
      int m = half ? (r + 8) : r;
      bilin[(b0 + m) * P_DIM + p] = v;
    }
  }
}

// out[b] = sigmoid( sum_p tanh(bilin + [le,re].V_ntn[p] + b_ntn[p]) * w_fc[p] + b_fc )
__global__ void final_head(const float* __restrict__ leS, const float* __restrict__ reS,
                           const float* __restrict__ bilin,
                           const float* __restrict__ V_ntn, const float* __restrict__ b_ntn,
                           const float* __restrict__ w_fc, const float* __restrict__ b_fc,
                           float* __restrict__ out) {
  int b = blockIdx.x * blockDim.x + threadIdx.x;
  if (b >= B_DIM) return;
  float s = 0.f;
  for (int pp = 0; pp < P_DIM; ++pp) {
    float acc = bilin[b * P_DIM + pp] + b_ntn[pp];
    const float* vr = V_ntn + pp * 256;
    #pragma unroll 4
    for (int c = 0; c < H; ++c)
      acc += leS[b * H + c] * vr[c] + reS[b * H + c] * vr[H + c];
    s += tanhf(acc) * w_fc[pp];
  }
  out[b] = 1.f / (1.f + expf(-(s + b_fc[0])));
}

// ------------------------------------------------------------------
extern "C" void kernel_launch(void* const* d_in, const int* in_sizes, int n_in,
                              void* d_out, int out_size, void* d_ws, size_t ws_size,
                              hipStream_t stream) {
  (void)in_sizes; (void)n_in; (void)out_size; (void)ws_size;
  const int*   left_x   = (const int*)d_in[0];
  const int*   right_x  = (const int*)d_in[2];
  const int*   left_b   = (const int*)d_in[4];
  const int*   right_b  = (const int*)d_in[5];
  const int*   anc_d    = (const int*)d_in[8];
  const int*   leaf_d   = (const int*)d_in[9];
  const int*   anc_p    = (const int*)d_in[10];
  const int*   leaf_p   = (const int*)d_in[11];
  const int*   anc_a    = (const int*)d_in[12];
  const int*   leaf_a   = (const int*)d_in[13];
  const float* emb_d    = (const float*)d_in[14];
  const float* emb_p    = (const float*)d_in[15];
  const float* emb_a    = (const float*)d_in[16];
  const float* Wl_d     = (const float*)d_in[17];
  const float* bl_d     = (const float*)d_in[18];
  const float* ap_d     = (const float*)d_in[19];
  const float* Wl_p     = (const float*)d_in[20];
  const float* bl_p     = (const float*)d_in[21];
  const float* ap_p     = (const float*)d_in[22];
  const float* Wl_a     = (const float*)d_in[23];
  const float* bl_a     = (const float*)d_in[24];
  const float* ap_a     = (const float*)d_in[25];
  const float* W_ntn    = (const float*)d_in[26];
  const float* V_ntn    = (const float*)d_in[27];
  const float* b_ntn    = (const float*)d_in[28];
  const float* w_fc     = (const float*)d_in[29];
  const float* b_fc     = (const float*)d_in[30];

  float* ws      = (float*)d_ws;
  float* all_emb = ws + OFF_ALLEMB;
  float* leS     = ws + OFF_LE;
  float* reS     = ws + OFF_RE;
  float* wsum    = ws + OFF_W;
  float* aw      = ws + OFF_AW;
  float* Wt      = ws + OFF_WT;
  float* bilin   = ws + OFF_BILIN;
  float* Wlpk    = ws + OFF_WLPK;
  float* out     = (float*)d_out;

  // zero le + re + wsum + aw (contiguous): graph-replay safe
  zero_f32<<<(NZERO + 255) / 256, 256, 0, stream>>>(leS, NZERO);

  // weight packing (K-pair layouts for b64 WMMA B loads)
  pack_wl<<<(32768 + 255) / 256, 256, 0, stream>>>(Wl_d, Wlpk);
  pack_wl<<<(32768 + 255) / 256, 256, 0, stream>>>(Wl_p, Wlpk + 32768);
  pack_wl<<<(32768 + 255) / 256, 256, 0, stream>>>(Wl_a, Wlpk + 65536);
  pack_wntn<<<(262144 + 255) / 256, 256, 0, stream>>>(W_ntn, Wt);

  // group d: V=10000, L=4 -> 40000 rows, 2500 tiles
  attn_wmma<<<dim3(2500, 8), 32, 0, stream>>>(anc_d, leaf_d, emb_d, Wlpk, bl_d, ap_d, aw);
  attn_softmax<<<(10000 + 127) / 128, 128, 0, stream>>>(aw, wsum, 10000, 4);
  emb_combine<<<10000, 128, 0, stream>>>(anc_d, emb_d, wsum, all_emb, 4);
  // group p: V=4000, L=4 -> 16000 rows, 1000 tiles
  attn_wmma<<<dim3(1000, 8), 32, 0, stream>>>(anc_p, leaf_p, emb_p, Wlpk + 32768, bl_p, ap_p,
                                              aw + 40000);
  attn_softmax<<<(4000 + 127) / 128, 128, 0, stream>>>(aw + 40000, wsum + 8, 4000, 4);
  emb_combine<<<4000, 128, 0, stream>>>(anc_p, emb_p, wsum + 8, all_emb + 10000 * H, 4);
  // group a: V=4000, L=5 -> 20000 rows, 1250 tiles
  attn_wmma<<<dim3(1250, 8), 32, 0, stream>>>(anc_a, leaf_a, emb_a, Wlpk + 65536, bl_a, ap_a,
                                              aw + 56000);
  attn_softmax<<<(4000 + 127) / 128, 128, 0, stream>>>(aw + 56000, wsum + 16, 4000, 5);
  emb_combine<<<4000, 128, 0, stream>>>(anc_a, emb_a, wsum + 16, all_emb + 14000 * H, 5);

  // segment sums (sorted batch ids)
  seg_sum<<<T_DIM / 64, 128, 0, stream>>>(left_x, left_b, all_emb, leS);
  seg_sum<<<T_DIM / 64, 128, 0, stream>>>(right_x, right_b, all_emb, reS);

  // bilinear NTN via fp32 WMMA
  bilinear_wmma<<<dim3(B_DIM / 16, P_DIM), 32, 0, stream>>>(leS, reS, Wt, bilin);

  // head
  final_head<<<(B_DIM + 127) / 128, 128, 0, stream>>>(leS, reS, bilin, V_ntn, b_ntn,
                                                      w_fc, b_fc, out);
}